// PhononUnfolding_80204219286222
// MI455X (gfx1250) — compile-verified
//
#include <hip/hip_runtime.h>

typedef float v2f __attribute__((ext_vector_type(2)));
typedef float v8f __attribute__((ext_vector_type(8)));

#define NQ 128
#define NM 128
#define ND 48
#define NG_ 64
#define NAK 48          // 3 * nkappa(16)
#define AK_PER_CHUNK 8
#define NCHUNK 6        // 48 / 8
#define P_SCALE (1.0f / 128.0f)   // Omega / tilde_Omega / ng = 1/2/64
#define RTOL_ 1e-5f
#define ATOL_ 1e-8f

// Phase 2: per (q-point i, ak-chunk) compute partial
//   Pp[i][chunk][m] = sum_{ak in chunk} sum_g mask[i,g] * (sum_d A[m,d]*B[d,g])^2
// A[m,d] = e_tilde[a,kap,m,i,d], B[d,g] = e[a,kap,d,g,i]
__global__ __launch_bounds__(256)
void PhononUnfolding_gemm_kernel(const float* __restrict__ q,
                                 const float* __restrict__ et,   // e_tilde
                                 const float* __restrict__ e,
                                 const float* __restrict__ gv,
                                 const float* __restrict__ G,
                                 const float* __restrict__ Q,
                                 float* __restrict__ Pp)          // [NQ][NCHUNK][NM]
{
    const int i     = blockIdx.x;
    const int chunk = blockIdx.y;
    const int t     = threadIdx.x;
    const int lane  = t & 31;
    const int wave  = t >> 5;        // 0..7 -> m-tile index
    const int l15   = lane & 15;
    const int hi    = lane >> 4;     // 0: lanes 0-15, 1: lanes 16-31

    __shared__ float maskf[NG_];
    if (t < NG_) {
        float ux = q[i*3+0] + gv[t*3+0] - G[0];
        float uy = q[i*3+1] + gv[t*3+1] - G[1];
        float uz = q[i*3+2] + gv[t*3+2] - G[2];
        bool ok = (fabsf(Q[i*3+0] - ux) <= ATOL_ + RTOL_ * fabsf(ux)) &&
                  (fabsf(Q[i*3+1] - uy) <= ATOL_ + RTOL_ * fabsf(uy)) &&
                  (fabsf(Q[i*3+2] - uz) <= ATOL_ + RTOL_ * fabsf(uz));
        maskf[t] = ok ? 1.0f : 0.0f;
    }
    __syncthreads();

    float m4[4];
#pragma unroll
    for (int gt = 0; gt < 4; ++gt) m4[gt] = maskf[gt*16 + l15];

    const int mLane = wave*16 + l15;     // row of A this lane supplies

    float acc[8];
#pragma unroll
    for (int r = 0; r < 8; ++r) acc[r] = 0.0f;

    for (int j = 0; j < AK_PER_CHUNK; ++j) {
        const int ak = chunk*AK_PER_CHUNK + j;

        // ---- A fragments: e_tilde[(ak*NM + m)*NQ*ND + i*ND + d] ----
        // ISA 16x4 f32 A layout: lanes 0-15 hold K=0,1 ; lanes 16-31 hold K=2,3
        const float* Ab = et + ((size_t)ak*NM + (size_t)mLane)*(NQ*ND)
                             + (size_t)i*ND + (hi << 1);
        // prefetch next (a,kappa) A row while we compute this one
        if (j + 1 < AK_PER_CHUNK) {
            __builtin_prefetch(et + ((size_t)(ak+1)*NM + (size_t)mLane)*(NQ*ND)
                                  + (size_t)i*ND, 0, 0);
        }
        v2f a[12];
#pragma unroll
        for (int kk = 0; kk < 12; ++kk)
            a[kk] = *(const v2f*)(Ab + 4*kk);

        // ---- B: e[(ak*ND + d)*NG*NQ + g*NQ + i] ----
        const float* Bb = e + (size_t)ak*(ND*NG_*NQ) + (size_t)i;

#pragma unroll
        for (int gt = 0; gt < 4; ++gt) {
            const int g = gt*16 + l15;   // column of B this lane supplies
            v8f c = {0.f,0.f,0.f,0.f,0.f,0.f,0.f,0.f};
#pragma unroll
            for (int kk = 0; kk < 12; ++kk) {
                const int dd = 4*kk + (hi << 1);   // K=0/1 (lo half) or 2/3 (hi half)
                v2f b;
                b.x = Bb[((size_t)dd      * NG_ + (size_t)g) * NQ];
                b.y = Bb[((size_t)(dd+1)  * NG_ + (size_t)g) * NQ];
                c = __builtin_amdgcn_wmma_f32_16x16x4_f32(
                        /*neg_a=*/false, a[kk], /*neg_b=*/false, b,
                        /*c_mod=*/(short)0, c, /*reuse_a=*/false, /*reuse_b=*/false);
            }
            // D tile layout: VGPR r, lanes 0-15 -> M=r ; lanes 16-31 -> M=r+8 ; N = lane&15
#pragma unroll
            for (int r = 0; r < 8; ++r)
                acc[r] = fmaf(m4[gt] * c[r], c[r], acc[r]);
        }
    }

    // Reduce over g (the 16 lanes within each half-wave)
#pragma unroll
    for (int r = 0; r < 8; ++r) {
#pragma unroll
        for (int off = 1; off < 16; off <<= 1)
            acc[r] += __shfl_xor(acc[r], off, 16);
    }

    if (l15 == 0) {
        // rows M=0..7 live in half 0 (VGPR r -> M=r), rows 8..15 in half 1
        float* dst = Pp + ((size_t)i*NCHUNK + chunk)*NM + wave*16 + hi*8;
#pragma unroll
        for (int r = 0; r < 8; ++r) dst[r] = acc[r];
    }
}

// Phase 3: P[k,m] = scale * sum_chunk Pp[k][c][m];
//          A_mu[m] = sum_k (omega[k]==omega[m]) * P[k,m]; broadcast to (NQ, NM)
__global__ __launch_bounds__(128)
void PhononUnfolding_finalize_kernel(const float* __restrict__ Pp,
                                     const float* __restrict__ omega,
                                     float* __restrict__ out)
{
    const int m = threadIdx.x;          // 0..127
    const float wm = omega[m];
    float amu = 0.0f;
    for (int k = 0; k < NM; ++k) {
        if (omega[k] == wm) {
            float p = 0.0f;
#pragma unroll
            for (int cc = 0; cc < NCHUNK; ++cc)
                p += Pp[((size_t)k*NCHUNK + cc)*NM + m];
            amu += p;
        }
    }
    amu *= P_SCALE;
    for (int i = 0; i < NQ; ++i)
        out[(size_t)i*NM + m] = amu;
}

extern "C" void kernel_launch(void* const* d_in, const int* in_sizes, int n_in,
                              void* d_out, int out_size, void* d_ws, size_t ws_size,
                              hipStream_t stream) {
    const float* q  = (const float*)d_in[0];   // (128,3)
    const float* om = (const float*)d_in[1];   // (128,)
    const float* et = (const float*)d_in[2];   // (3,16,128,128,48)
    const float* e  = (const float*)d_in[3];   // (3,16,48,64,128)
    const float* gv = (const float*)d_in[4];   // (64,3)
    const float* G  = (const float*)d_in[5];   // (3,)
    const float* Q  = (const float*)d_in[6];   // (128,3)

    float* Pp = (float*)d_ws;   // NQ*NCHUNK*NM floats = 393216 bytes; fully overwritten

    PhononUnfolding_gemm_kernel<<<dim3(NQ, NCHUNK), 256, 0, stream>>>(
        q, et, e, gv, G, Q, Pp);
    PhononUnfolding_finalize_kernel<<<1, NM, 0, stream>>>(
        Pp, om, (float*)d_out);
}